// VectorQuantizer_51642686767885
// MI455X (gfx1250) — compile-verified
//
#include <hip/hip_runtime.h>
#include <hip/hip_bf16.h>
#include <math.h>

typedef __attribute__((ext_vector_type(2))) float v2f;
typedef __attribute__((ext_vector_type(8))) float v8f;

#define NPTS   65536      // B*H*W
#define CDIM   64
#define KCODES 1024
#define CB_STRIDE 68      // padded LDS stride (floats) to avoid bank conflicts

// output layout (floats): q[4194304] | q_st[4194304] | indices[65536] | perplexity[1]
#define Q_OFF    0
#define QST_OFF  4194304
#define IDX_OFF  8388608
#define PERP_OFF 8454144

// ---------------- kernel 0: codebook norms + zero histogram ----------------
__global__ void vq_init_kernel(const float* __restrict__ cb,
                               float* __restrict__ eNorm,
                               unsigned* __restrict__ counts) {
    int k = blockIdx.x * blockDim.x + threadIdx.x;
    if (k < KCODES) {
        const float* row = cb + k * CDIM;
        float s = 0.f;
        #pragma unroll
        for (int c = 0; c < CDIM; ++c) s += row[c] * row[c];
        eNorm[k] = s;
        counts[k] = 0u;
    }
}

// ---------------- kernel 1: fused distances + argmin + gather ----------------
__launch_bounds__(256)
__global__ void vq_main_kernel(const float* __restrict__ z,
                               const float* __restrict__ cb,
                               const float* __restrict__ eNorm,
                               float* __restrict__ out,
                               unsigned* __restrict__ counts) {
    __shared__ float lds_cb[KCODES * CB_STRIDE];  // 278528 B
    __shared__ float lds_en[KCODES];              //   4096 B
    __shared__ int   lds_idx[8 * 16];             //    512 B

    const int tid  = threadIdx.x;
    const int lane = tid & 31;
    const int wave = tid >> 5;
    const int m    = lane & 15;
    const bool hi  = lane >= 16;

    // ---- stage full codebook (float4) + eNorm into LDS ----
    const float4* cb4 = (const float4*)cb;
    for (int i = tid; i < KCODES * CDIM / 4; i += 256) {
        float4 v = cb4[i];
        int row = i >> 4;
        int c4  = (i & 15) << 2;
        *(float4*)&lds_cb[row * CB_STRIDE + c4] = v;
    }
    for (int i = tid; i < KCODES; i += 256) lds_en[i] = eNorm[i];
    __syncthreads();

    // ---- load 16x64 A tile into registers (WMMA f32 16x16x4 A layout) ----
    const int mtile  = blockIdx.x * 8 + wave;
    const int base_n = mtile * 16;
    const int n      = base_n + m;
    const int rowbase = (n >> 10) * (CDIM * 1024) + (n & 1023);  // z[b, c, hw]
    v2f a[16];
    #pragma unroll
    for (int cc = 0; cc < 16; ++cc) {
        int c0 = cc * 4 + (hi ? 2 : 0);
        a[cc].x = z[rowbase + c0 * 1024];
        a[cc].y = z[rowbase + (c0 + 1) * 1024];
    }

    float minv[8];
    int   mini[8];
    #pragma unroll
    for (int v = 0; v < 8; ++v) { minv[v] = 3.0e38f; mini[v] = 0; }

    // ---- sweep all codes in 16-wide N tiles ----
    for (int t = 0; t < KCODES / 16; ++t) {
        const int j  = t * 16 + m;                 // this lane's code column
        const int jb = j * CB_STRIDE + (hi ? 2 : 0);
        v8f acc0 = {};
        v8f acc1 = {};
        #pragma unroll
        for (int cc = 0; cc < 16; cc += 2) {
            v2f b0 = *(const v2f*)&lds_cb[jb + 4 * cc];
            v2f b1 = *(const v2f*)&lds_cb[jb + 4 * (cc + 1)];
            acc0 = __builtin_amdgcn_wmma_f32_16x16x4_f32(
                false, a[cc],     false, b0, (short)0, acc0, false, false);
            acc1 = __builtin_amdgcn_wmma_f32_16x16x4_f32(
                false, a[cc + 1], false, b1, (short)0, acc1, false, false);
        }
        const float en = lds_en[j];
        #pragma unroll
        for (int v = 0; v < 8; ++v) {
            float d = en - 2.0f * (acc0[v] + acc1[v]);  // ||e||^2 - 2 x.e
            if (d < minv[v]) { minv[v] = d; mini[v] = j; }
        }
    }

    // ---- cross-lane argmin butterfly (within each 16-lane half) ----
    #pragma unroll
    for (int v = 0; v < 8; ++v) {
        float bv = minv[v];
        int   bi = mini[v];
        #pragma unroll
        for (int s = 1; s <= 8; s <<= 1) {
            float ov = __shfl_xor(bv, s, 32);
            int   oi = __shfl_xor(bi, s, 32);
            if (ov < bv || (ov == bv && oi < bi)) { bv = ov; bi = oi; }
        }
        minv[v] = bv; mini[v] = bi;
    }
    // lane 0 holds results for points M=0..7, lane 16 for M=8..15
    if (m == 0) {
        #pragma unroll
        for (int v = 0; v < 8; ++v)
            lds_idx[wave * 16 + v + (hi ? 8 : 0)] = mini[v];
    }
    // per-wave LDS slot: no cross-wave race; dscnt wait inserted by compiler
    const int jIdx = lds_idx[wave * 16 + m];

    // ---- outputs: indices (float), histogram, gathered quantized tensors ----
    if (!hi) {
        out[IDX_OFF + base_n + m] = (float)jIdx;
        atomicAdd(&counts[jIdx], 1u);
    }
    const int jrow = jIdx * CB_STRIDE;
    #pragma unroll 8
    for (int k2 = 0; k2 < 32; ++k2) {
        int c = 2 * k2 + (hi ? 1 : 0);
        float v = lds_cb[jrow + c];
        int o = rowbase + c * 1024;            // [B,C,H,W] layout, coalesced in n
        out[Q_OFF + o]   = v;
        out[QST_OFF + o] = v;                  // z + stopgrad(q-z) == q in value
    }
}

// ---------------- kernel 2: perplexity ----------------
__global__ void vq_perp_kernel(const unsigned* __restrict__ counts,
                               float* __restrict__ out) {
    __shared__ float red[1024];
    int t = threadIdx.x;
    float p = (float)counts[t] * (1.0f / (float)NPTS);
    red[t] = p * logf(fmaxf(p, 1e-10f));
    __syncthreads();
    for (int s = 512; s > 0; s >>= 1) {
        if (t < s) red[t] += red[t + s];
        __syncthreads();
    }
    if (t == 0) out[PERP_OFF] = expf(-red[0]);
}

extern "C" void kernel_launch(void* const* d_in, const int* in_sizes, int n_in,
                              void* d_out, int out_size, void* d_ws, size_t ws_size,
                              hipStream_t stream) {
    const float* z  = (const float*)d_in[0];   // [64,64,32,32]
    const float* cb = (const float*)d_in[1];   // [1024,64]
    float* out = (float*)d_out;

    float*    ws_en     = (float*)d_ws;                 // 1024 floats
    unsigned* ws_counts = (unsigned*)((char*)d_ws + KCODES * sizeof(float));

    vq_init_kernel<<<4, 256, 0, stream>>>(cb, ws_en, ws_counts);
    vq_main_kernel<<<NPTS / 128, 256, 0, stream>>>(z, cb, ws_en, out, ws_counts);
    vq_perp_kernel<<<1, 1024, 0, stream>>>(ws_counts, out);
}